// StreamingHyperbolicAttention_16870631539369
// MI455X (gfx1250) — compile-verified
//
#include <hip/hip_runtime.h>
#include <math.h>

typedef __attribute__((ext_vector_type(2))) float v2f;
typedef __attribute__((ext_vector_type(8))) float v8f;

#define D_MODEL 1024
#define N_SEQ   2048
#define BATCH   2
#define HEADS   16
#define HDIM    64
#define BN      (BATCH * N_SEQ)   // 4096
#define CURV    1.0f

// D = A(16x4, f32) * B(4x16, f32) + C(16x16, f32)
#define WMMA_F32(a, b, c) \
  __builtin_amdgcn_wmma_f32_16x16x4_f32(false, (a), false, (b), (short)0, (c), false, false)

// LDS row stride (floats): 68 -> rows hit banks 4*r mod 64, conflict-free for
// 16-lane half-wave reads; keeps b64/b128 LDS alignment.
#define LSTR 68

// ---------------------------------------------------------------------------
// GEMM: Y[m][n] = sum_k X[m][k] * W[n][k]  (Y = X @ W^T), M=BN, N=K=1024.
// Block = 8 waves sharing one 64-column n-chunk; W K-chunks (64x64) are staged
// through LDS once per block (register-pipelined), each wave computes its own
// 16x64 tile with A operands streamed from global.
// ---------------------------------------------------------------------------
__global__ __launch_bounds__(256) void gemm_xwt(const float* __restrict__ X,
                                                const float* __restrict__ W,
                                                float* __restrict__ Y)
{
  __shared__ __align__(16) float wslab[64 * LSTR];   // 17.4 KB

  const int tid  = threadIdx.x;
  const int winb = tid >> 5;
  const int lane = tid & 31;
  const int half = lane >> 4;
  const int l16  = lane & 15;

  const int nblocks = D_MODEL / 64;          // 16 n-chunks
  const int nc = blockIdx.x % nblocks;
  const int mg = blockIdx.x / nblocks;       // 32 m-groups of 128 rows
  const int mt = mg * 8 + winb;              // this wave's 16-row m-tile

  const float* xrow = X + (size_t)(mt * 16 + l16) * D_MODEL + 2 * half;

  // cooperative W staging: thread covers rows wr, wr+16, wr+32, wr+48
  const int wr  = tid >> 4;                  // 0..15
  const int wc4 = (tid & 15) * 4;            // col within 64-wide K-chunk
  const float* wbase = W + (size_t)(nc * 64 + wr) * D_MODEL + wc4;

  float4 stage[4];
  #pragma unroll
  for (int rr = 0; rr < 4; ++rr)
    stage[rr] = *(const float4*)(wbase + (size_t)(rr * 16) * D_MODEL);

  v8f acc0 = {}, acc1 = {}, acc2 = {}, acc3 = {};

  for (int kk = 0; kk < D_MODEL; kk += 64) {
    __syncthreads();                                   // LDS consumed
    #pragma unroll
    for (int rr = 0; rr < 4; ++rr)
      *(float4*)(wslab + (wr + rr * 16) * LSTR + wc4) = stage[rr];
    __syncthreads();                                   // LDS ready

    if (kk + 64 < D_MODEL) {                           // prefetch next chunk
      #pragma unroll
      for (int rr = 0; rr < 4; ++rr)
        stage[rr] = *(const float4*)(wbase + (size_t)(rr * 16) * D_MODEL + kk + 64);
    }

    #pragma unroll
    for (int j = 0; j < 16; ++j) {                     // 16 K-steps of 4
      v2f a  = *(const v2f*)(xrow + kk + j * 4);
      v2f b0 = *(const v2f*)(wslab + ( 0 + l16) * LSTR + j * 4 + 2 * half);
      v2f b1 = *(const v2f*)(wslab + (16 + l16) * LSTR + j * 4 + 2 * half);
      v2f b2 = *(const v2f*)(wslab + (32 + l16) * LSTR + j * 4 + 2 * half);
      v2f b3 = *(const v2f*)(wslab + (48 + l16) * LSTR + j * 4 + 2 * half);
      acc0 = WMMA_F32(a, b0, acc0);
      acc1 = WMMA_F32(a, b1, acc1);
      acc2 = WMMA_F32(a, b2, acc2);
      acc3 = WMMA_F32(a, b3, acc3);
    }
  }

  // C layout: VGPR v holds row M = v + 8*half, col N = l16
  #pragma unroll
  for (int v = 0; v < 8; ++v) {
    float* yr = Y + (size_t)(mt * 16 + v + 8 * half) * D_MODEL + nc * 64 + l16;
    yr[0]  = acc0[v];
    yr[16] = acc1[v];
    yr[32] = acc2[v];
    yr[48] = acc3[v];
  }
}

// ---------------------------------------------------------------------------
// Streaming hyperbolic attention. Block = 8 waves on 8 consecutive q-tiles of
// one (b,h); each 16x64 K and V tile is staged into LDS once per block and
// shared. Scores via f32 WMMA, arccosh-distance + online softmax on the
// C-layout tile, P restaged via LDS to A layout, P@V via WMMA.
// ---------------------------------------------------------------------------
__global__ __launch_bounds__(256) void hyp_attn(const float* __restrict__ Q,
                                                const float* __restrict__ K,
                                                const float* __restrict__ V,
                                                float* __restrict__ O)
{
  __shared__ __align__(16) float kslab[16 * LSTR];     // 4.4 KB
  __shared__ __align__(16) float vslab[16 * LSTR];     // 4.4 KB
  __shared__ __align__(16) float pst[8 * 16 * 18];     // per-wave P staging

  const int tid  = threadIdx.x;
  const int winb = tid >> 5;
  const int lane = tid & 31;
  const int half = lane >> 4;
  const int l16  = lane & 15;
  float* pslab = pst + winb * (16 * 18);

  const int qg = blockIdx.x & 15;            // q-tile group (16 per head)
  const int h  = (blockIdx.x >> 4) & 15;
  const int b  = blockIdx.x >> 8;
  const int qt = qg * 8 + winb;
  const int r0 = b * N_SEQ + qt * 16;        // first query row (global)
  const int cb = h * HDIM;                   // head column base

  // K/V cooperative staging indices: one float4 per tile per thread
  const int sr  = tid >> 4;                  // row 0..15
  const int sc4 = (tid & 15) * 4;            // col 0..60

  // ---- preload 16x64 Q tile in A layout; q norms for free ----
  v2f aq[16];
  float qsum = 0.f;
  const float* qbase = Q + (size_t)(r0 + l16) * D_MODEL + cb + 2 * half;
  #pragma unroll
  for (int j = 0; j < 16; ++j) {
    aq[j] = *(const v2f*)(qbase + j * 4);
    qsum += aq[j].x * aq[j].x + aq[j].y * aq[j].y;
  }
  qsum += __shfl_xor(qsum, 16);              // 64-wide row sums at lanes m, m+16
  float qn8[8];
  #pragma unroll
  for (int v = 0; v < 8; ++v)
    qn8[v] = __shfl(qsum, v + 8 * half);     // qn for row M = v + 8*half

  float m_run[8], l_run[8];
  #pragma unroll
  for (int v = 0; v < 8; ++v) { m_run[v] = -3.0e38f; l_run[v] = 0.f; }
  v8f o0 = {}, o1 = {}, o2 = {}, o3 = {};

  const int NT = N_SEQ / 16;
  for (int kt = 0; kt < NT; ++kt) {
    const int k0 = b * N_SEQ + kt * 16;

    // ---- stage K/V tiles into LDS (once per block) ----
    __syncthreads();
    *(float4*)(kslab + sr * LSTR + sc4) =
        *(const float4*)(K + (size_t)(k0 + sr) * D_MODEL + cb + sc4);
    *(float4*)(vslab + sr * LSTR + sc4) =
        *(const float4*)(V + (size_t)(k0 + sr) * D_MODEL + cb + sc4);
    __syncthreads();

    if (kt + 1 < NT) {                       // CDNA5 global_prefetch of next tiles
      __builtin_prefetch(K + (size_t)(k0 + 16 + sr) * D_MODEL + cb + sc4, 0, 1);
      __builtin_prefetch(V + (size_t)(k0 + 16 + sr) * D_MODEL + cb + sc4, 0, 1);
    }

    // ---- K tile in B layout from LDS; k norms ----
    v2f bk[16];
    float ksum = 0.f;
    #pragma unroll
    for (int j = 0; j < 16; ++j) {
      bk[j] = *(const v2f*)(kslab + l16 * LSTR + j * 4 + 2 * half);
      ksum += bk[j].x * bk[j].x + bk[j].y * bk[j].y;
    }
    ksum += __shfl_xor(ksum, 16);            // kn for key col N = l16

    // ---- S = Q K^T (16 f32 WMMAs over K=64) ----
    v8f s = {};
    #pragma unroll
    for (int j = 0; j < 16; ++j)
      s = WMMA_F32(aq[j], bk[j], s);

    // ---- hyperbolic distance + online softmax on C-layout tile ----
    float p[8], scl[8];
    #pragma unroll
    for (int v = 0; v < 8; ++v) {
      const float qn = qn8[v];
      const float kn = ksum;
      float d2   = fmaxf(qn + kn - 2.f * s[v], 0.f);
      float den  = fmaxf((1.f - CURV * qn) * (1.f - CURV * kn), 1e-6f);
      float xarg = fmaxf(1.f + 2.f * CURV * d2 / den, 1.f);
      float dist = logf(xarg + sqrtf(xarg * xarg - 1.f));   // arccosh
      float sc   = -dist;

      float mv = sc;                         // row max within 16-lane half
      mv = fmaxf(mv, __shfl_xor(mv, 1));
      mv = fmaxf(mv, __shfl_xor(mv, 2));
      mv = fmaxf(mv, __shfl_xor(mv, 4));
      mv = fmaxf(mv, __shfl_xor(mv, 8));
      float mnew = fmaxf(m_run[v], mv);
      float sf   = __expf(m_run[v] - mnew);  // 0 on first tile
      m_run[v] = mnew;

      float pv = __expf(sc - mnew);
      float rs = pv;
      rs += __shfl_xor(rs, 1);
      rs += __shfl_xor(rs, 2);
      rs += __shfl_xor(rs, 4);
      rs += __shfl_xor(rs, 8);
      l_run[v] = l_run[v] * sf + rs;
      p[v] = pv; scl[v] = sf;
    }

    // ---- stage P (C layout -> LDS row-major, per-wave slab) ----
    #pragma unroll
    for (int v = 0; v < 8; ++v)
      pslab[(v + 8 * half) * 18 + l16] = p[v];

    // ---- rescale running output ----
    #pragma unroll
    for (int v = 0; v < 8; ++v) {
      o0[v] *= scl[v]; o1[v] *= scl[v]; o2[v] *= scl[v]; o3[v] *= scl[v];
    }

    asm volatile("s_wait_dscnt 0" ::: "memory");   // same-wave LDS RAW fence

    // ---- O += P @ V : K=16 in 4 steps, 4 N-chunks of 16, V from LDS ----
    #pragma unroll
    for (int js = 0; js < 4; ++js) {
      v2f ap = *(const v2f*)(pslab + l16 * 18 + js * 4 + 2 * half);
      const float* vr0 = vslab + (js * 4 + 2 * half) * LSTR + l16;
      const float* vr1 = vr0 + LSTR;
      v2f b0 = { vr0[0],  vr1[0]  };
      v2f b1 = { vr0[16], vr1[16] };
      v2f b2 = { vr0[32], vr1[32] };
      v2f b3 = { vr0[48], vr1[48] };
      o0 = WMMA_F32(ap, b0, o0);
      o1 = WMMA_F32(ap, b1, o1);
      o2 = WMMA_F32(ap, b2, o2);
      o3 = WMMA_F32(ap, b3, o3);
    }
  }

  // ---- normalize and store (merged-head layout [BN, D]) ----
  #pragma unroll
  for (int v = 0; v < 8; ++v) {
    float inv = 1.0f / l_run[v];
    float* orow = O + (size_t)(r0 + v + 8 * half) * D_MODEL + cb + l16;
    orow[0]  = o0[v] * inv;
    orow[16] = o1[v] * inv;
    orow[32] = o2[v] * inv;
    orow[48] = o3[v] * inv;
  }
}

// ---------------------------------------------------------------------------
extern "C" void kernel_launch(void* const* d_in, const int* in_sizes, int n_in,
                              void* d_out, int out_size, void* d_ws, size_t ws_size,
                              hipStream_t stream) {
  const float* x  = (const float*)d_in[0];
  const float* wq = (const float*)d_in[1];
  const float* wk = (const float*)d_in[2];
  const float* wv = (const float*)d_in[3];
  const float* wo = (const float*)d_in[4];
  float* out = (float*)d_out;

  float* ws = (float*)d_ws;
  const size_t MAT = (size_t)BN * D_MODEL;   // 4M floats = 16 MB
  float* q  = ws;
  float* k  = ws + MAT;
  float* v  = ws + 2 * MAT;
  float* ao = ws + 3 * MAT;

  dim3 blk(256);
  dim3 gemm_grid((BN / 128) * (D_MODEL / 64));         // 32 * 16 = 512 blocks
  gemm_xwt<<<gemm_grid, blk, 0, stream>>>(x, wq, q);
  gemm_xwt<<<gemm_grid, blk, 0, stream>>>(x, wk, k);
  gemm_xwt<<<gemm_grid, blk, 0, stream>>>(x, wv, v);

  dim3 attn_grid(BATCH * HEADS * (N_SEQ / 128));       // 512 blocks, 8 q-tiles each
  hyp_attn<<<attn_grid, blk, 0, stream>>>(q, k, v, ao);

  gemm_xwt<<<gemm_grid, blk, 0, stream>>>(ao, wo, out);
}